// UniGCNII_70909910057320
// MI455X (gfx1250) — compile-verified
//
#include <hip/hip_runtime.h>
#include <math.h>

// ---------------------------------------------------------------------------
// UniGCNII forward for MI455X (gfx1250, wave32).
// GEMMs use V_WMMA_F32_16X16X4_F32 (native fp32 matrix path, exact fp32 FMA).
// Scatter-mean stages are HBM-bound; implemented with float atomics.
// ---------------------------------------------------------------------------

typedef float v2f __attribute__((ext_vector_type(2)));
typedef float v8f __attribute__((ext_vector_type(8)));

#define NNODES 50000
#define NEDGES 10000
#define NNZ_   800000
#define FDIM   128
#define FOUT   64
#define NLAYER 2

// ---------------- zero fill (grid-stride) ----------------
__global__ void zero_f32(float* __restrict__ p, long n) {
  long t = (long)blockIdx.x * blockDim.x + threadIdx.x;
  long stride = (long)gridDim.x * blockDim.x;
  for (; t < n; t += stride) p[t] = 0.0f;
}

// ---------------- count occurrences ----------------
__global__ void count_kernel(const int* __restrict__ idx, float* __restrict__ cnt, int nnz) {
  int t = blockIdx.x * blockDim.x + threadIdx.x;
  if (t < nnz) atomicAdd(&cnt[idx[t]], 1.0f);
}

// ---------------- fused gather + segment-sum: dst[sidx[nz]] += src[gidx[nz]] ----------------
__global__ void scatter_add_128(const float* __restrict__ src, const int* __restrict__ gidx,
                                const int* __restrict__ sidx, float* __restrict__ dst, int nnz) {
  long t = (long)blockIdx.x * blockDim.x + threadIdx.x;
  long total = (long)nnz * FDIM;
  if (t >= total) return;
  int nz = (int)(t >> 7);
  int f  = (int)(t & 127);
  float v = src[(long)gidx[nz] * FDIM + f];
  atomicAdd(&dst[(long)sidx[nz] * FDIM + f], v);
}

// ---------------- mean: out = sum / max(cnt,1) ----------------
__global__ void divide_128(const float* __restrict__ sum, const float* __restrict__ cnt,
                           float* __restrict__ out, int nrows) {
  long t = (long)blockIdx.x * blockDim.x + threadIdx.x;
  long total = (long)nrows * FDIM;
  if (t >= total) return;
  int r = (int)(t >> 7);
  out[t] = sum[t] / fmaxf(cnt[r], 1.0f);
}

// ---------------- Xv = vsum/cnt; l2-normalize row; Xi = 0.9*X + 0.1*x0 ----------------
__global__ void norm_combine(const float* __restrict__ vsum, const float* __restrict__ vcnt,
                             const float* __restrict__ x0, float* __restrict__ xi) {
  __shared__ float red[4];
  int row = blockIdx.x;
  int f = threadIdx.x;                 // 128 threads = 4 wave32
  long base = (long)row * FDIM;
  float c = fmaxf(vcnt[row], 1.0f);
  float xv = vsum[base + f] / c;
  float s = xv * xv;
  #pragma unroll
  for (int off = 16; off > 0; off >>= 1) s += __shfl_down(s, off, 32);
  int wid = f >> 5, lid = f & 31;
  if (lid == 0) red[wid] = s;
  __syncthreads();
  float tot = red[0] + red[1] + red[2] + red[3];
  float norm = sqrtf(tot);
  float scale = (norm > 0.0f) ? (1.0f / fmaxf(norm, 1e-30f)) : 0.0f;
  xi[base + f] = 0.9f * xv * scale + 0.1f * x0[base + f];
}

// ---------------- fp32 WMMA GEMM: C = epi(alpha*(A@W) + bias + sbeta*skip) ----------------
// A: [M,K] row-major, W: [K,N] row-major, C/skip: [M,N].
// blockDim.x = 32*(N/16); wave w owns output tile (blockIdx.x*16 .. +16) x (w*16 .. +16).
// Requires M % 16 == 0 (holds: M = 50000).
__global__ void gemm_wmma_f32(const float* __restrict__ A, const float* __restrict__ W,
                              const float* __restrict__ bias, const float* __restrict__ skip,
                              float alpha, float sbeta, int relu,
                              float* __restrict__ C, int M, int K, int N) {
  int wave = threadIdx.x >> 5;
  int lane = threadIdx.x & 31;
  int half = lane >> 4;      // 0: lanes 0-15, 1: lanes 16-31
  int l16  = lane & 15;
  int m0 = blockIdx.x * 16;
  int n0 = wave * 16;
  long arow = (long)(m0 + l16) * K;    // A row for this lane (M = l16)

  v8f acc = {};
  for (int k0 = 0; k0 < K; k0 += 4) {
    int ka = k0 + 2 * half;
    // A 16x4 tile: lane l16 = row, holds K = {ka, ka+1}
    v2f a;
    a.x = A[arow + ka];
    a.y = A[arow + ka + 1];
    // B 4x16 tile: VGPR r, half h -> row K = k0 + 2h + r, col N = n0 + l16
    v2f b;
    b.x = W[(long)(ka    ) * N + n0 + l16];
    b.y = W[(long)(ka + 1) * N + n0 + l16];
    acc = __builtin_amdgcn_wmma_f32_16x16x4_f32(false, a, false, b,
                                                (short)0, acc, false, false);
  }

  int ncol = n0 + l16;
  float bv = bias ? bias[ncol] : 0.0f;
  #pragma unroll
  for (int r = 0; r < 8; ++r) {
    int m = m0 + r + 8 * half;         // C/D layout: VGPR r, half h -> M = r + 8h
    float v = alpha * acc[r] + bv;
    if (skip) v += sbeta * skip[(long)m * N + ncol];
    if (relu) v = fmaxf(v, 0.0f);
    C[(long)m * N + ncol] = v;
  }
}

// ---------------------------------------------------------------------------
extern "C" void kernel_launch(void* const* d_in, const int* in_sizes, int n_in,
                              void* d_out, int out_size, void* d_ws, size_t ws_size,
                              hipStream_t stream) {
  const float* x_in   = (const float*)d_in[0];   // [50000,128]
  const float* W0     = (const float*)d_in[1];   // [128,128]
  const float* b0     = (const float*)d_in[2];   // [128]
  const float* Ws     = (const float*)d_in[3];   // [2,128,128]
  const float* Wout   = (const float*)d_in[4];   // [128,64]
  const float* bout   = (const float*)d_in[5];   // [64]
  const int*   vertex = (const int*)d_in[6];     // [800000]
  const int*   edges  = (const int*)d_in[7];     // [800000]
  float* out = (float*)d_out;                    // [50000*64] ++ [10000*128]

  // workspace carve-up (floats)
  float* ws   = (float*)d_ws;
  float* x0   = ws; ws += (long)NNODES * FDIM;   // relu(x@W0+b0), kept as residual
  float* xA   = ws; ws += (long)NNODES * FDIM;   // per-layer activation
  float* xi   = ws; ws += (long)NNODES * FDIM;   // Xi buffer (GEMM input)
  float* vsum = ws; ws += (long)NNODES * FDIM;
  float* esum = ws; ws += (long)NEDGES * FDIM;
  float* xe   = ws; ws += (long)NEDGES * FDIM;
  float* ecnt = ws; ws += NEDGES;
  float* vcnt = ws; ws += NNODES;

  // incidence counts (fixed across layers)
  zero_f32<<<64, 256, 0, stream>>>(ecnt, NEDGES);
  zero_f32<<<256, 256, 0, stream>>>(vcnt, NNODES);
  count_kernel<<<(NNZ_ + 255) / 256, 256, 0, stream>>>(edges, ecnt, NNZ_);
  count_kernel<<<(NNZ_ + 255) / 256, 256, 0, stream>>>(vertex, vcnt, NNZ_);

  // input linear + relu -> x0
  gemm_wmma_f32<<<NNODES / 16, 32 * (FDIM / 16), 0, stream>>>(
      x_in, W0, b0, nullptr, 1.0f, 0.0f, 1, x0, NNODES, FDIM, FDIM);

  const long scat_blocks = ((long)NNZ_ * FDIM) / 256;
  const float* xcur = x0;
  for (int i = 0; i < NLAYER; ++i) {
    float beta = logf(0.5f / (float)(i + 1) + 1.0f);
    // last layer's Xe is the second tuple output -> write it straight into d_out
    float* xeDst = (i == NLAYER - 1) ? (out + (long)NNODES * FOUT) : xe;

    zero_f32<<<1024, 256, 0, stream>>>(esum, (long)NEDGES * FDIM);
    scatter_add_128<<<(int)scat_blocks, 256, 0, stream>>>(xcur, vertex, edges, esum, NNZ_);
    divide_128<<<(int)(((long)NEDGES * FDIM + 255) / 256), 256, 0, stream>>>(esum, ecnt, xeDst, NEDGES);

    zero_f32<<<4096, 256, 0, stream>>>(vsum, (long)NNODES * FDIM);
    scatter_add_128<<<(int)scat_blocks, 256, 0, stream>>>(xeDst, edges, vertex, vsum, NNZ_);
    norm_combine<<<NNODES, FDIM, 0, stream>>>(vsum, vcnt, x0, xi);

    // x = relu((1-beta)*Xi + beta*(Xi @ Ws[i]))   (xA is not an input -> in-place safe)
    gemm_wmma_f32<<<NNODES / 16, 32 * (FDIM / 16), 0, stream>>>(
        xi, Ws + (long)i * FDIM * FDIM, nullptr, xi, beta, 1.0f - beta, 1,
        xA, NNODES, FDIM, FDIM);
    xcur = xA;
  }

  // output linear (no relu) -> first tuple output
  gemm_wmma_f32<<<NNODES / 16, 32 * (FOUT / 16), 0, stream>>>(
      xcur, Wout, bout, nullptr, 1.0f, 0.0f, 0, out, NNODES, FDIM, FOUT);
}